// ExpertRaceGate_20160576487589
// MI455X (gfx1250) — compile-verified
//
#include <hip/hip_runtime.h>
#include <hip/hip_bf16.h>

typedef __attribute__((ext_vector_type(2))) float v2f;
typedef __attribute__((ext_vector_type(8))) float v8f;

#define N_TOK   8192
#define EMB     4096
#define NEXP    64
#define TOTAL   (N_TOK * NEXP)   // 524288
#define KSEL    16384u           // int(N * 2.0)
#define NCHUNK  8
#define CHUNK   (N_TOK / NCHUNK) // 1024

// ---------------------------------------------------------------------------
// Kernel 1: logits[N,64] = H[N,4096] @ W[64,4096]^T  via V_WMMA_F32_16X16X4_F32
// Register-blocked: each wave computes 32 rows x 16 experts with TWO
// accumulators sharing one B fetch (halves L2 traffic vs 16x16/wave; A is
// read exactly once from HBM/L2, W re-reads drop to ~256MB total).
// Block = 256 threads = 8 waves covering 64 rows x 64 experts.
//   wave w: rows r0 = 64*bx + 32*(w>>2) (two 16-row subtiles), experts (w&3)*16
// ---------------------------------------------------------------------------
__global__ __launch_bounds__(256) void gemm_logits(const float* __restrict__ H,
                                                   const float* __restrict__ W,
                                                   float* __restrict__ L) {
    const int w    = threadIdx.x >> 5;
    const int lane = threadIdx.x & 31;
    const int half = lane >> 4;      // 0: K{0,1}, 1: K{2,3}
    const int l16  = lane & 15;
    const int r0   = blockIdx.x * 64 + (w >> 2) * 32;
    const int e0   = (w & 3) * 16;

    const float* ap0 = H + (size_t)(r0 + l16) * EMB + 2 * half;   // A rows [r0, r0+16)
    const float* ap1 = ap0 + (size_t)16 * EMB;                    // A rows [r0+16, r0+32)
    const float* bp  = W + (size_t)(e0 + l16) * EMB + 2 * half;   // B[k][e] = W[e][k]

    v8f acc0 = {};
    v8f acc1 = {};
#pragma unroll 4
    for (int k = 0; k < EMB; k += 4) {
        if ((k & 15) == 0 && k + 1024 < EMB) {
            // 4KB-ahead stream prefetch; locality 3 -> WGP scope (installs in
            // all cache levels per CDNA5 prefetch scope rules)
            __builtin_prefetch(ap0 + k + 1024, 0, 3);
            __builtin_prefetch(ap1 + k + 1024, 0, 3);
            __builtin_prefetch(bp + k + 1024, 0, 3);
        }
        v2f b  = *(const v2f*)(bp + k);
        v2f a0 = *(const v2f*)(ap0 + k);
        v2f a1 = *(const v2f*)(ap1 + k);
        acc0 = __builtin_amdgcn_wmma_f32_16x16x4_f32(false, a0, false, b,
                                                     (short)0, acc0, false, false);
        acc1 = __builtin_amdgcn_wmma_f32_16x16x4_f32(false, a1, false, b,
                                                     (short)0, acc1, false, false);
    }
    // C/D layout: VGPR v -> row (tile_r0) + v + 8*half, col e0 + l16
    float* out0 = L + (size_t)(r0 + 8 * half) * NEXP + e0 + l16;
    float* out1 = out0 + (size_t)16 * NEXP;
#pragma unroll
    for (int v = 0; v < 8; ++v) {
        out0[(size_t)v * NEXP] = acc0[v];
        out1[(size_t)v * NEXP] = acc1[v];
    }
}

// ---------------------------------------------------------------------------
// Radix select (4 passes, 8 bits each, descending) for the K-th largest value.
// Monotone key: u = (b & 0x80000000) ? ~b : (b | 0x80000000)
// state[0..255]=bins, state[256]=remaining rank, state[257]=key prefix.
// ---------------------------------------------------------------------------
__global__ void sel_init(unsigned* state) {
    int t = threadIdx.x;
    if (t < 256) state[t] = 0u;
    if (t == 0) { state[256] = KSEL; state[257] = 0u; }
}

__global__ __launch_bounds__(256) void sel_hist(const float* __restrict__ L,
                                                unsigned* __restrict__ state,
                                                int shift) {
    __shared__ unsigned lb[256];
    lb[threadIdx.x] = 0u;
    __syncthreads();
    const unsigned upfx = state[257];
    const int stride = gridDim.x * blockDim.x;
    for (int i = blockIdx.x * blockDim.x + threadIdx.x; i < TOTAL; i += stride) {
        unsigned b = __float_as_uint(L[i]);
        unsigned u = (b & 0x80000000u) ? ~b : (b | 0x80000000u);
        // match previously-selected high bytes (bits >= shift+8); shift+8 may be 32
        if ((((unsigned long long)(u ^ upfx)) >> (shift + 8)) == 0ull)
            atomicAdd(&lb[(u >> shift) & 255u], 1u);
    }
    __syncthreads();
    atomicAdd(&state[threadIdx.x], lb[threadIdx.x]);
}

__global__ __launch_bounds__(256) void sel_scan(unsigned* __restrict__ state,
                                                int shift, int last,
                                                float* __restrict__ kth) {
    __shared__ unsigned tmp[256];
    __shared__ int sel;
    const int t = threadIdx.x;
    const unsigned krem = state[256];
    const unsigned upfx = state[257];
    tmp[t] = state[t];
    if (t == 0) sel = 0;
    __syncthreads();
    state[t] = 0u;                       // reset bins for next pass
    // suffix (descending) inclusive scan: tmp[t] = sum_{b>=t} bins[b]
    for (int off = 1; off < 256; off <<= 1) {
        unsigned v = (t + off < 256) ? tmp[t + off] : 0u;
        __syncthreads();
        tmp[t] += v;
        __syncthreads();
    }
    if (tmp[t] >= krem) atomicMax(&sel, t);
    __syncthreads();
    if (t == 0) {
        int s = sel;
        unsigned cnt_gt = (s < 255) ? tmp[s + 1] : 0u; // elems with strictly larger digit
        state[256] = krem - cnt_gt;
        unsigned np = upfx | ((unsigned)s << shift);
        state[257] = np;
        if (last) {
            unsigned b = (np & 0x80000000u) ? (np ^ 0x80000000u) : ~np;
            *kth = __uint_as_float(b);
        }
    }
}

// ---------------------------------------------------------------------------
// Softmax (per row over 64 experts) + mask + final_weights.
// 8 waves per block, one wave32 per row, 2 columns per lane.
// ---------------------------------------------------------------------------
__global__ __launch_bounds__(256) void softmax_mask(const float* __restrict__ L,
                                                    const float* __restrict__ kth,
                                                    float* __restrict__ P,
                                                    float* __restrict__ M,
                                                    float* __restrict__ FW) {
    const int w = threadIdx.x >> 5;
    const int lane = threadIdx.x & 31;
    const int row = blockIdx.x * 8 + w;
    const float kv = *kth;
    const size_t off = (size_t)row * NEXP + 2 * lane;
    const float x0 = L[off], x1 = L[off + 1];

    float mx = fmaxf(x0, x1);
    for (int m = 16; m >= 1; m >>= 1) mx = fmaxf(mx, __shfl_xor(mx, m, 32));
    float e0 = __expf(x0 - mx), e1 = __expf(x1 - mx);
    float s = e0 + e1;
    for (int m = 16; m >= 1; m >>= 1) s += __shfl_xor(s, m, 32);
    const float inv = 1.0f / s;

    P[off] = e0 * inv;  P[off + 1] = e1 * inv;
    const float m0 = (x0 >= kv) ? 1.0f : 0.0f;
    const float m1 = (x1 >= kv) ? 1.0f : 0.0f;
    M[off] = m0;        M[off + 1] = m1;
    FW[off] = x0 * m0;  FW[off + 1] = x1 * m1;
}

// ---------------------------------------------------------------------------
// Gram matrices P'=P^T P and M'=M^T M (64x64) via WMMA, K-chunked.
// grid = (NCHUNK, 2); y=0 -> P, y=1 -> M. 8 waves, 16 tiles of 16x16 (2/wave).
// Each block writes a private partial (no FP atomics -> deterministic).
// ---------------------------------------------------------------------------
__global__ __launch_bounds__(256) void gram(const float* __restrict__ Pbuf,
                                            const float* __restrict__ Mbuf,
                                            float* __restrict__ part) {
    const float* S = blockIdx.y ? Mbuf : Pbuf;
    float* dst = part + ((size_t)blockIdx.y * NCHUNK + blockIdx.x) * (NEXP * NEXP);
    const int w = threadIdx.x >> 5;
    const int lane = threadIdx.x & 31;
    const int half = lane >> 4, l16 = lane & 15;
    const int k0 = blockIdx.x * CHUNK;

    for (int t = w; t < 16; t += 8) {
        const int i = (t >> 2) * 16, j = (t & 3) * 16;
        v8f acc = {};
        // A[e1,k] = S[k][e1] (transposed read), B[k,e2] = S[k][e2]
        const float* ap = S + (size_t)(k0 + 2 * half) * NEXP + i + l16;
        const float* bp = S + (size_t)(k0 + 2 * half) * NEXP + j + l16;
#pragma unroll 4
        for (int k = 0; k < CHUNK; k += 4) {
            v2f a, b;
            a[0] = ap[(size_t)k * NEXP];  a[1] = ap[(size_t)k * NEXP + NEXP];
            b[0] = bp[(size_t)k * NEXP];  b[1] = bp[(size_t)k * NEXP + NEXP];
            acc = __builtin_amdgcn_wmma_f32_16x16x4_f32(false, a, false, b,
                                                        (short)0, acc, false, false);
        }
        float* o = dst + (size_t)(i + 8 * half) * NEXP + j + l16;
#pragma unroll
        for (int v = 0; v < 8; ++v) o[(size_t)v * NEXP] = acc[v];
    }
}

// ---------------------------------------------------------------------------
// Final lsim: fold W-matrix algebra into 4 scalar sums over the 64x64 Grams.
// ---------------------------------------------------------------------------
__global__ __launch_bounds__(256) void finalize(const float* __restrict__ part,
                                                float* __restrict__ outScalar) {
    const float* pP = part;
    const float* pM = part + (size_t)NCHUNK * NEXP * NEXP;
    float s_all = 0.f, s_diag = 0.f, s_offp = 0.f, s_diagp = 0.f;
    for (int idx = threadIdx.x; idx < NEXP * NEXP; idx += 256) {
        float pp = 0.f, mm = 0.f;
#pragma unroll
        for (int c = 0; c < NCHUNK; ++c) {
            pp += pP[c * NEXP * NEXP + idx];
            mm += pM[c * NEXP * NEXP + idx];
        }
        const int i = idx >> 6, j = idx & 63;
        s_all += mm;
        if (i == j) { s_diag += mm; s_diagp += mm * pp; }
        else        { s_offp += mm * pp; }
    }
    __shared__ float red[256];
    float sums[4] = { s_all, s_diag, s_offp, s_diagp };
    float out[4];
    for (int r = 0; r < 4; ++r) {
        red[threadIdx.x] = sums[r];
        __syncthreads();
        for (int o = 128; o > 0; o >>= 1) {
            if (threadIdx.x < o) red[threadIdx.x] += red[threadIdx.x + o];
            __syncthreads();
        }
        out[r] = red[0];
        __syncthreads();
    }
    if (threadIdx.x == 0) {
        const float eps = 1.1920928955078125e-7f;   // float32 eps
        const float E = (float)NEXP;
        const float off_factor = (E * E - E) / (out[0] + eps);
        const float lsim = (off_factor * out[2] + (E / (out[1] + eps)) * out[3]) / E;
        *outScalar = lsim;
    }
}

// ---------------------------------------------------------------------------
extern "C" void kernel_launch(void* const* d_in, const int* in_sizes, int n_in,
                              void* d_out, int out_size, void* d_ws, size_t ws_size,
                              hipStream_t stream) {
    const float* H = (const float*)d_in[0];   // [8192, 4096]
    const float* W = (const float*)d_in[1];   // [64, 4096]
    float* FW = (float*)d_out;                // [8192*64] final_weights
    float* lsim_out = (float*)d_out + TOTAL;  // [1] lsim

    float* logits = (float*)d_ws;                         // 524288
    float* Pbuf   = logits + TOTAL;                       // 524288
    float* Mbuf   = Pbuf + TOTAL;                         // 524288
    float* part   = Mbuf + TOTAL;                         // 2*8*4096
    unsigned* state = (unsigned*)(part + 2 * NCHUNK * NEXP * NEXP); // 260
    float* kth    = (float*)(state + 260);                // 1

    gemm_logits<<<N_TOK / 64, 256, 0, stream>>>(H, W, logits);

    sel_init<<<1, 256, 0, stream>>>(state);
    for (int p = 0; p < 4; ++p) {
        const int shift = 24 - 8 * p;
        sel_hist<<<512, 256, 0, stream>>>(logits, state, shift);
        sel_scan<<<1, 256, 0, stream>>>(state, shift, (p == 3) ? 1 : 0, kth);
    }

    softmax_mask<<<N_TOK / 8, 256, 0, stream>>>(logits, kth, Pbuf, Mbuf, FW);
    gram<<<dim3(NCHUNK, 2), 256, 0, stream>>>(Pbuf, Mbuf, part);
    finalize<<<1, 256, 0, stream>>>(part, lsim_out);
}